// SimilarityGate_54236847014129
// MI455X (gfx1250) — compile-verified
//
#include <hip/hip_runtime.h>
#include <hip/hip_bf16.h>
#include <math.h>

typedef __attribute__((ext_vector_type(2))) float v2f;
typedef __attribute__((ext_vector_type(8))) float v8f;

#define MODEL_DIM   2048
#define NUM_HEADS   8
#define NUM_EXPERTS 256
#define SQRT_E      16
#define SUBKEY_K    4
#define FINAL_K     16
#define NUM_SELECTED 128          // NUM_HEADS * FINAL_K
#define N_TOK       2048
#define D_HEAD      256           // per-side half of each head's 512-dim q
#define CAPACITY    128
#define CDIM        ((size_t)33554432)  // 2048*128*128

// ---------------------------------------------------------------------------
// Kernel A1: fold sub-keys into W_effT[col][k], col = s*128 + h*16 + e
//   W_effT[col, k] = sum_d sk_s[e, d] * w_q[h*512+s*256+d, k]
// WMMA f32 16x16x4: M = expert e (16), N = k tile (16), K = d (256).
// A (sk rows) is contiguous -> v2f b64 loads; B (w_q rows) is lane-coalesced.
// ---------------------------------------------------------------------------
__global__ void build_weff(const float* __restrict__ w_q,
                           const float* __restrict__ sk1,
                           const float* __restrict__ sk2,
                           float* __restrict__ weffT) {
    int wg   = blockIdx.x * 8 + (threadIdx.x >> 5);  // 16 cb x 128 kt = 2048
    int lane = threadIdx.x & 31;
    int kt = wg & 127;           // k tile (N dim)
    int cb = wg >> 7;            // col block = s*8 + h
    int s  = cb >> 3, h = cb & 7;
    const float* sk = s ? sk2 : sk1;
    int rowbase = h * 512 + s * 256;     // row offset into w_q
    int r16   = lane & 15;
    int khalf = lane >> 4;
    int k0 = kt * 16;
    v8f c = {};
    for (int d = 0; d < D_HEAD; d += 4) {
        int dd = d + 2 * khalf;
        v2f a = *(const v2f*)(sk + r16 * D_HEAD + dd);       // A[m=e][dd..dd+1]
        v2f b;                                               // B[dd][n=k]
        b.x = w_q[(size_t)(rowbase + dd)     * MODEL_DIM + k0 + r16];
        b.y = w_q[(size_t)(rowbase + dd + 1) * MODEL_DIM + k0 + r16];
        c = __builtin_amdgcn_wmma_f32_16x16x4_f32(false, a, false, b,
                                                  (short)0, c, false, false);
    }
    int col0 = s * 128 + h * 16;
#pragma unroll
    for (int r = 0; r < 8; ++r) {        // D: lane n=k, VGPR r -> e row
        int col = col0 + r + 8 * khalf;
        weffT[(size_t)col * MODEL_DIM + k0 + r16] = c[r];
    }
}

// ---------------------------------------------------------------------------
// Kernel A2: scores (2048 x 256) = x (2048 x 2048) @ W_effT^T
// Both A and B fragments are per-lane contiguous v2f -> global_load_b64 only.
// ---------------------------------------------------------------------------
__global__ void score_gemm(const float* __restrict__ x,
                           const float* __restrict__ weffT,
                           float* __restrict__ scores) {
    int wg   = blockIdx.x * 8 + (threadIdx.x >> 5);  // 128 x 16 = 2048 tiles
    int lane = threadIdx.x & 31;
    int mt = wg >> 4;           // token tile
    int nt = wg & 15;           // col tile
    int r16   = lane & 15;
    int khalf = lane >> 4;
    int m  = mt * 16 + r16;
    int n0 = nt * 16;
    const float* arow = x     + (size_t)m * MODEL_DIM;
    const float* brow = weffT + (size_t)(n0 + r16) * MODEL_DIM;
    v8f c = {};
    for (int k = 0; k < MODEL_DIM; k += 4) {
        int kk = k + 2 * khalf;
        v2f a = *(const v2f*)(arow + kk);   // A[m][kk..kk+1], 8B aligned
        v2f b = *(const v2f*)(brow + kk);   // B[kk..kk+1][n] via transposed Weff
        c = __builtin_amdgcn_wmma_f32_16x16x4_f32(false, a, false, b,
                                                  (short)0, c, false, false);
    }
#pragma unroll
    for (int r = 0; r < 8; ++r) {
        scores[(size_t)(mt * 16 + r + 8 * khalf) * NUM_EXPERTS + n0 + r16] = c[r];
    }
}

// ---------------------------------------------------------------------------
// Kernel B: per-token gating. One wave32 block per token.
// Threads 0..7 handle heads: stable top-4 per side, 4x4 combine, stable full
// descending sort (lax.top_k tie rule: lower index first). Then softmax over
// the 128 slots, emit gates, expert ids, per-token 256-bin histogram.
// ---------------------------------------------------------------------------
__global__ void gate_kernel(const float* __restrict__ scores,
                            float* __restrict__ gates,
                            int*   __restrict__ flat_idx,
                            float* __restrict__ flat_idx_out,
                            int*   __restrict__ hist) {
    int t   = blockIdx.x;
    int tid = threadIdx.x;   // 0..31
    __shared__ float vals[NUM_SELECTED];
    __shared__ int   eids[NUM_SELECTED];
    __shared__ int   lhist[NUM_EXPERTS];
    __shared__ float ssum;

    if (tid < NUM_HEADS) {
        int h = tid;
        float s1[SQRT_E], s2[SQRT_E];
        for (int e = 0; e < SQRT_E; ++e) {
            s1[e] = scores[(size_t)t * NUM_EXPERTS +       h * 16 + e];
            s2[e] = scores[(size_t)t * NUM_EXPERTS + 128 + h * 16 + e];
        }
        float ts1[SUBKEY_K], ts2[SUBKEY_K];
        int   ti1[SUBKEY_K], ti2[SUBKEY_K];
        bool u1[SQRT_E], u2[SQRT_E];
        for (int e = 0; e < SQRT_E; ++e) { u1[e] = false; u2[e] = false; }
        for (int r = 0; r < SUBKEY_K; ++r) {            // stable top-4
            int b1 = -1, b2 = -1; float v1 = -INFINITY, v2v = -INFINITY;
            for (int e = 0; e < SQRT_E; ++e) {
                if (!u1[e] && s1[e] > v1)  { v1  = s1[e]; b1 = e; }
                if (!u2[e] && s2[e] > v2v) { v2v = s2[e]; b2 = e; }
            }
            u1[b1] = true; u2[b2] = true;
            ts1[r] = v1; ti1[r] = b1; ts2[r] = v2v; ti2[r] = b2;
        }
        float comb[16];
        for (int i = 0; i < SUBKEY_K; ++i)
            for (int j = 0; j < SUBKEY_K; ++j)
                comb[i * SUBKEY_K + j] = ts1[i] + ts2[j];
        bool used[16];
        for (int f = 0; f < 16; ++f) used[f] = false;
        for (int r = 0; r < FINAL_K; ++r) {             // stable full sort desc
            int best = 0; float bv = -INFINITY;
            for (int f = 0; f < 16; ++f)
                if (!used[f] && comb[f] > bv) { bv = comb[f]; best = f; }
            used[best] = true;
            vals[h * FINAL_K + r] = bv;
            eids[h * FINAL_K + r] = ti1[best >> 2] * SQRT_E + ti2[best & 3];
        }
    }
    for (int i = tid; i < NUM_EXPERTS; i += 32) lhist[i] = 0;
    __syncthreads();
    if (tid == 0) {                                     // softmax (serial, 128)
        float mx = -INFINITY;
        for (int i = 0; i < NUM_SELECTED; ++i) mx = fmaxf(mx, vals[i]);
        float sum = 0.f;
        for (int i = 0; i < NUM_SELECTED; ++i) {
            float e = expf(vals[i] - mx); vals[i] = e; sum += e;
        }
        ssum = sum;
    }
    __syncthreads();
    float inv = 1.0f / ssum;
    for (int i = tid; i < NUM_SELECTED; i += 32) {
        float g = vals[i] * inv;
        int   e = eids[i];
        size_t p = (size_t)t * NUM_SELECTED + i;
        gates[p] = g;
        flat_idx[p] = e;
        flat_idx_out[p] = (float)e;
        atomicAdd(&lhist[e], 1);
    }
    __syncthreads();
    for (int i = tid; i < NUM_EXPERTS; i += 32)
        hist[(size_t)t * NUM_EXPERTS + i] = lhist[i];
}

// ---------------------------------------------------------------------------
// Kernel C: per-expert deterministic reduction -> exp_counts, me*ce
// ---------------------------------------------------------------------------
__global__ void expert_stats(const int* __restrict__ flat_idx,
                             const float* __restrict__ gates,
                             float* __restrict__ exp_counts_out,
                             float* __restrict__ mece) {
    int e = blockIdx.x, tid = threadIdx.x;
    const int total = N_TOK * NUM_SELECTED;
    int cnt = 0; float gs = 0.f;
    for (int i = tid; i < total; i += 256)
        if (flat_idx[i] == e) { cnt++; gs += gates[i]; }
    __shared__ int   sc[256];
    __shared__ float sg[256];
    sc[tid] = cnt; sg[tid] = gs; __syncthreads();
    for (int off = 128; off > 0; off >>= 1) {
        if (tid < off) { sc[tid] += sc[tid + off]; sg[tid] += sg[tid + off]; }
        __syncthreads();
    }
    if (tid == 0) {
        float me = sg[0] / (float)N_TOK;
        float ce = (float)sc[0] / (float)(N_TOK * NUM_SELECTED);
        exp_counts_out[e] = (float)sc[0];
        mece[e] = me * ce;
    }
}

__global__ void laux_kernel(const float* __restrict__ mece,
                            float* __restrict__ out0) {
    __shared__ float s[NUM_EXPERTS];
    int tid = threadIdx.x;
    s[tid] = mece[tid];
    __syncthreads();
    for (int off = 128; off > 0; off >>= 1) {
        if (tid < off) s[tid] += s[tid + off];
        __syncthreads();
    }
    // mean(me*ce) * E * E / NUM_SELECTED = (sum/256) * 256*256/128 = sum*2
    if (tid == 0) out0[0] = s[0] * 2.0f;
}

// ---------------------------------------------------------------------------
// Kernel D: per-expert serial prefix over token histograms -> base ranks.
// Reproduces the stable argsort rank exactly (token-major flat order).
// ---------------------------------------------------------------------------
__global__ void prefix_kernel(const int* __restrict__ hist,
                              int* __restrict__ offs) {
    int e = threadIdx.x;   // 256 threads, 1 block
    int run = 0;
    for (int t = 0; t < N_TOK; ++t) {
        offs[(size_t)t * NUM_EXPERTS + e] = run;
        run += hist[(size_t)t * NUM_EXPERTS + e];
    }
}

// ---------------------------------------------------------------------------
// Kernel E: per-token scatter of combine_weights / dispatch_mask nonzeros.
// ---------------------------------------------------------------------------
__global__ void scatter_kernel(const int* __restrict__ flat_idx,
                               const float* __restrict__ gates,
                               const int* __restrict__ offs,
                               float* __restrict__ combine,
                               float* __restrict__ dispatch) {
    int t = blockIdx.x;
    int s = threadIdx.x;   // 0..127 slots
    __shared__ int   eid[NUM_SELECTED];
    __shared__ float sred[NUM_SELECTED];
    size_t p = (size_t)t * NUM_SELECTED + s;
    int   e  = flat_idx[p];
    float gv = gates[p];
    eid[s] = e;
    __syncthreads();
    int rank = offs[(size_t)t * NUM_EXPERTS + e];
    for (int s2 = 0; s2 < s; ++s2) if (eid[s2] == e) rank++;
    bool mask = rank < CAPACITY;
    float gm  = mask ? gv : 0.f;
    sred[s] = gm; __syncthreads();
    for (int off = 64; off > 0; off >>= 1) {
        if (s < off) sred[s] += sred[s + off];
        __syncthreads();
    }
    float denom = fmaxf(sred[0], 1.1920929e-07f);   // clip to f32 eps
    if (mask) {
        float gn = gm / denom;
        size_t idx = (size_t)t * (NUM_SELECTED * CAPACITY)
                   + (size_t)s * CAPACITY + rank;
        combine[idx]  = gn;
        dispatch[idx] = 1.0f;
    }
}

// ---------------------------------------------------------------------------
extern "C" void kernel_launch(void* const* d_in, const int* in_sizes, int n_in,
                              void* d_out, int out_size, void* d_ws, size_t ws_size,
                              hipStream_t stream) {
    (void)in_sizes; (void)n_in; (void)out_size; (void)ws_size;
    const float* x   = (const float*)d_in[0];
    const float* w_q = (const float*)d_in[1];
    const float* sk1 = (const float*)d_in[2];
    const float* sk2 = (const float*)d_in[3];

    char* ws = (char*)d_ws;
    float* weffT  = (float*)(ws);                 // 2 MB  (256 x 2048, K-major)
    float* scores = (float*)(ws + (2u << 20));    // 2 MB  (2048 x 256)
    float* gates  = (float*)(ws + (4u << 20));    // 1 MB  (2048 x 128)
    int*   fidx   = (int*)  (ws + (5u << 20));    // 1 MB
    int*   hist   = (int*)  (ws + (6u << 20));    // 2 MB  (2048 x 256)
    int*   offs   = (int*)  (ws + (8u << 20));    // 2 MB
    float* mece   = (float*)(ws + (10u << 20));   // 1 KB

    float* out          = (float*)d_out;
    float* out_laux     = out;                      // [1]
    float* out_combine  = out + 1;                  // [2048,128,128]
    float* out_dispatch = out + 1 + CDIM;           // [2048,128,128]
    float* out_counts   = out + 1 + 2 * CDIM;       // [256]
    float* out_fidx     = out_counts + NUM_EXPERTS; // [2048,128]

    // zero the dense combine/dispatch region (268 MB) — dominant HBM cost
    hipMemsetAsync(out_combine, 0, 2 * CDIM * sizeof(float), stream);

    build_weff    <<<256, 256, 0, stream>>>(w_q, sk1, sk2, weffT);
    score_gemm    <<<256, 256, 0, stream>>>(x, weffT, scores);
    gate_kernel   <<<N_TOK, 32, 0, stream>>>(scores, gates, fidx, out_fidx, hist);
    expert_stats  <<<NUM_EXPERTS, 256, 0, stream>>>(fidx, gates, out_counts, mece);
    laux_kernel   <<<1, NUM_EXPERTS, 0, stream>>>(mece, out_laux);
    prefix_kernel <<<1, NUM_EXPERTS, 0, stream>>>(hist, offs);
    scatter_kernel<<<N_TOK, NUM_SELECTED, 0, stream>>>(fidx, gates, offs,
                                                       out_combine, out_dispatch);
}